// GCN_21844203668000
// MI455X (gfx1250) — compile-verified
//
#include <hip/hip_runtime.h>
#include <math.h>

#define F_IN 16
#define HID  64
#define NCLS 2
#define GG   256
#define EPSV 1e-5f

typedef __attribute__((ext_vector_type(2))) float v2f;
typedef __attribute__((ext_vector_type(8))) float v8f;

static inline int cdiv_h(int a, int b) { return (a + b - 1) / b; }

// ---------------- elementwise / utility kernels ----------------

__global__ void k_fill(float* __restrict__ p, float v, int n) {
    int t = blockIdx.x * blockDim.x + threadIdx.x;
    if (t < n) p[t] = v;
}

__global__ void k_deg(const int* __restrict__ row, float* __restrict__ deg, int E) {
    int t = blockIdx.x * blockDim.x + threadIdx.x;
    if (t < E) atomicAdd(&deg[row[t]], 1.0f);
}

__global__ void k_rsqrt(float* __restrict__ p, int n) {
    int t = blockIdx.x * blockDim.x + threadIdx.x;
    if (t < n) p[t] = rsqrtf(p[t]);
}

__global__ void k_cnt(const int* __restrict__ batch, float* __restrict__ cnt, int n) {
    int t = blockIdx.x * blockDim.x + threadIdx.x;
    if (t < n) atomicAdd(&cnt[batch[t]], 1.0f);
}

// ---------------- GraphNorm kernels ----------------

__global__ void k_seg_sum(const float* __restrict__ src, const int* __restrict__ batch,
                          float* __restrict__ gsum, int n, int C) {
    int t = blockIdx.x * blockDim.x + threadIdx.x;
    if (t >= n * C) return;
    int node = t / C, c = t - node * C;
    atomicAdd(&gsum[batch[node] * C + c], src[t]);
}

__global__ void k_mean(const float* __restrict__ gsum, const float* __restrict__ cnt,
                       float* __restrict__ gmean, int G, int C) {
    int t = blockIdx.x * blockDim.x + threadIdx.x;
    if (t >= G * C) return;
    gmean[t] = gsum[t] / fmaxf(cnt[t / C], 1.0f);
}

__global__ void k_center_var(const float* __restrict__ src, float* __restrict__ dst,
                             const int* __restrict__ batch, const float* __restrict__ gmean,
                             const float* __restrict__ ms, float* __restrict__ gvarsum,
                             int n, int C) {
    int t = blockIdx.x * blockDim.x + threadIdx.x;
    if (t >= n * C) return;
    int node = t / C, c = t - node * C;
    int b = batch[node];
    float v = src[t] - ms[c] * gmean[b * C + c];
    dst[t] = v;
    atomicAdd(&gvarsum[b * C + c], v * v);
}

__global__ void k_norm_fin(float* __restrict__ h, const int* __restrict__ batch,
                           const float* __restrict__ gvar, const float* __restrict__ w,
                           const float* __restrict__ b, int n, int C) {
    int t = blockIdx.x * blockDim.x + threadIdx.x;
    if (t >= n * C) return;
    int node = t / C, c = t - node * C;
    h[t] = w[c] * h[t] * rsqrtf(gvar[batch[node] * C + c] + EPSV) + b[c];
}

// ---------------- WMMA fp32 GEMM: out[M,64] = A[M,K] @ W[K,64] (+bias)(+relu) ----------------
// 8 waves/block, each wave produces 16 rows x 64 cols via V_WMMA_F32_16X16X4_F32.
// Weights are staged in LDS pre-swizzled into B-fragment order so that each
// fragment is one aligned ds_load_b64:
//   lWp[((k>>1)*64 + n)*2 + (k&1)] = W[k][n]
// A lane's B fragment for (k0, half, col) is then the v2f at
//   lWp + (((k0>>1) + half)*64 + col) * 2

__global__ __launch_bounds__(256) void k_wmma_gemm(const float* __restrict__ A,
                                                   const float* __restrict__ W,
                                                   const float* __restrict__ bias,
                                                   float* __restrict__ out,
                                                   int M, int K, int do_relu) {
    __shared__ float lWp[HID * HID];  // up to 64x64 weights, fragment-swizzled
    int tid = threadIdx.x;
    for (int i = tid; i < K * HID; i += 256) {
        int k = i >> 6;          // i / 64
        int n = i & 63;          // i % 64
        lWp[(((k >> 1) * HID + n) << 1) + (k & 1)] = W[i];
    }
    __syncthreads();

    int wave = tid >> 5;
    int lane = tid & 31;
    int half = lane >> 4;   // 0: K pair {0,1} / M rows 0..7 ; 1: K pair {2,3} / M rows 8..15
    int nl   = lane & 15;   // M index for A-frag, N index for B/D-frags

    int rowBase = blockIdx.x * 128 + wave * 16;
    int mA = rowBase + nl;
    if (mA > M - 1) mA = M - 1;             // clamp so every lane loads valid data
    const float* arow = A + (size_t)mA * K;

    v8f acc0 = {}, acc1 = {}, acc2 = {}, acc3 = {};

    for (int k0 = 0; k0 < K; k0 += 4) {
        int ka = k0 + 2 * half;
        v2f a = *(const v2f*)&arow[ka];                       // global_load_b64
        const float* bbase = &lWp[(((k0 >> 1) + half) * HID + nl) << 1];
        v2f b0 = *(const v2f*)(bbase + 0);                    // ds_load_b64 (x4, mergeable)
        v2f b1 = *(const v2f*)(bbase + 32);
        v2f b2 = *(const v2f*)(bbase + 64);
        v2f b3 = *(const v2f*)(bbase + 96);
        acc0 = __builtin_amdgcn_wmma_f32_16x16x4_f32(false, a, false, b0, (short)0, acc0, false, false);
        acc1 = __builtin_amdgcn_wmma_f32_16x16x4_f32(false, a, false, b1, (short)0, acc1, false, false);
        acc2 = __builtin_amdgcn_wmma_f32_16x16x4_f32(false, a, false, b2, (short)0, acc2, false, false);
        acc3 = __builtin_amdgcn_wmma_f32_16x16x4_f32(false, a, false, b3, (short)0, acc3, false, false);
    }

    // D layout: VGPR r -> row (r + 8*half), lane nl -> column within 16-wide tile
    for (int r = 0; r < 8; r++) {
        int mrow = rowBase + r + 8 * half;
        if (mrow < M) {
            float v0 = acc0[r], v1 = acc1[r], v2 = acc2[r], v3 = acc3[r];
            if (bias) {
                v0 += bias[nl];
                v1 += bias[16 + nl];
                v2 += bias[32 + nl];
                v3 += bias[48 + nl];
            }
            if (do_relu) {
                v0 = fmaxf(v0, 0.0f); v1 = fmaxf(v1, 0.0f);
                v2 = fmaxf(v2, 0.0f); v3 = fmaxf(v3, 0.0f);
            }
            float* orow = out + (size_t)mrow * HID;
            orow[nl]      = v0;
            orow[16 + nl] = v1;
            orow[32 + nl] = v2;
            orow[48 + nl] = v3;
        }
    }
}

// ---------------- GCN edge aggregation ----------------
// 16 threads per edge, float4 per thread; gathers + atomics stay in the 192MB L2
// (h and agg are 25.6MB each).

__global__ void k_edge(const int* __restrict__ ei, const float* __restrict__ dinv,
                       const float* __restrict__ h, float* __restrict__ agg, int E) {
    int t = blockIdx.x * blockDim.x + threadIdx.x;
    int e = t >> 4;
    if (e >= E) return;
    int ci = (t & 15) << 2;
    int r = ei[e];
    int c = ei[E + e];
    float nrm = dinv[r] * dinv[c];
    const float4 v = *(const float4*)&h[(size_t)c * HID + ci];
    float* o = &agg[(size_t)r * HID + ci];
    atomicAdd(o + 0, nrm * v.x);
    atomicAdd(o + 1, nrm * v.y);
    atomicAdd(o + 2, nrm * v.z);
    atomicAdd(o + 3, nrm * v.w);
}

__global__ void k_self_relu(const float* __restrict__ agg, const float* __restrict__ h2,
                            const float* __restrict__ dinv, const float* __restrict__ bias,
                            float* __restrict__ out, int n) {
    int t = blockIdx.x * blockDim.x + threadIdx.x;
    if (t >= n * HID) return;
    int node = t / HID, c = t - node * HID;
    float di = dinv[node];
    out[t] = fmaxf(agg[t] + di * di * h2[t] + bias[c], 0.0f);
}

// ---------------- classifier head: logits + softmax ----------------

__global__ void k_head(const float* __restrict__ g, const float* __restrict__ Wo,
                       const float* __restrict__ bo, float* __restrict__ out, int G) {
    int i = blockIdx.x * blockDim.x + threadIdx.x;
    if (i >= G) return;
    float l0 = bo[0], l1 = bo[1];
    const float* gr = g + (size_t)i * HID;
    for (int k = 0; k < HID; k++) {
        float v = gr[k];
        l0 += v * Wo[k * NCLS + 0];
        l1 += v * Wo[k * NCLS + 1];
    }
    float m = fmaxf(l0, l1);
    float e0 = expf(l0 - m), e1 = expf(l1 - m);
    float s = e0 + e1;
    out[i * NCLS + 0] = e0 / s;
    out[i * NCLS + 1] = e1 / s;
}

// ---------------- launch ----------------

extern "C" void kernel_launch(void* const* d_in, const int* in_sizes, int n_in,
                              void* d_out, int out_size, void* d_ws, size_t ws_size,
                              hipStream_t stream) {
    const float* x      = (const float*)d_in[0];
    const int*   ei     = (const int*)d_in[1];
    const int*   batch  = (const int*)d_in[2];
    const float* gn_w[3]  = {(const float*)d_in[3],  (const float*)d_in[8],  (const float*)d_in[13]};
    const float* gn_b[3]  = {(const float*)d_in[4],  (const float*)d_in[9],  (const float*)d_in[14]};
    const float* gn_ms[3] = {(const float*)d_in[5],  (const float*)d_in[10], (const float*)d_in[15]};
    const float* Wc[3]    = {(const float*)d_in[6],  (const float*)d_in[11], (const float*)d_in[16]};
    const float* bc[3]    = {(const float*)d_in[7],  (const float*)d_in[12], (const float*)d_in[17]};
    const float* Wd = (const float*)d_in[18];
    const float* bd = (const float*)d_in[19];
    const float* Wo = (const float*)d_in[20];
    const float* bo = (const float*)d_in[21];

    const int N = in_sizes[0] / F_IN;
    const int E = in_sizes[1] / 2;
    const int G = GG;

    float* ws = (float*)d_ws;
    size_t o = 0;
    float* deg    = ws + o; o += (size_t)N;            // becomes dinv after rsqrt
    float* B0     = ws + o; o += (size_t)N * HID;      // normalized features / layer output
    float* B1     = ws + o; o += (size_t)N * HID;      // GEMM output h2
    float* B2     = ws + o; o += (size_t)N * HID;      // edge aggregation
    float* gcnt   = ws + o; o += (size_t)G;
    float* gsum   = ws + o; o += (size_t)G * HID;      // reused for mean-sum and var-sum
    float* gmean  = ws + o; o += (size_t)G * HID;
    float* gvar   = ws + o; o += (size_t)G * HID;
    float* gpool  = ws + o; o += (size_t)G * HID;
    float* gdense = ws + o; o += (size_t)G * HID;

    auto gs = [](int n) { return (n + 255) / 256; };

    // symmetric-norm degree terms (shared by all three convs)
    k_fill<<<gs(N), 256, 0, stream>>>(deg, 1.0f, N);       // +1 self-loop
    k_deg<<<gs(E), 256, 0, stream>>>(ei, deg, E);
    k_rsqrt<<<gs(N), 256, 0, stream>>>(deg, N);            // deg -> dinv
    // per-graph node counts (shared by all GraphNorms + pooling)
    k_fill<<<gs(G), 256, 0, stream>>>(gcnt, 0.0f, G);
    k_cnt<<<gs(N), 256, 0, stream>>>(batch, gcnt, N);

    const float* hsrc = x;
    int C = F_IN;
    for (int layer = 0; layer < 3; layer++) {
        // ---- GraphNorm(hsrc [N,C]) -> B0 [N,C]
        k_fill<<<gs(G * C), 256, 0, stream>>>(gsum, 0.0f, G * C);
        k_seg_sum<<<gs(N * C), 256, 0, stream>>>(hsrc, batch, gsum, N, C);
        k_mean<<<gs(G * C), 256, 0, stream>>>(gsum, gcnt, gmean, G, C);
        k_fill<<<gs(G * C), 256, 0, stream>>>(gsum, 0.0f, G * C);
        k_center_var<<<gs(N * C), 256, 0, stream>>>(hsrc, B0, batch, gmean, gn_ms[layer], gsum, N, C);
        k_mean<<<gs(G * C), 256, 0, stream>>>(gsum, gcnt, gvar, G, C);
        k_norm_fin<<<gs(N * C), 256, 0, stream>>>(B0, batch, gvar, gn_w[layer], gn_b[layer], N, C);
        // ---- h2 = B0 @ W  (WMMA fp32)
        k_wmma_gemm<<<cdiv_h(N, 128), 256, 0, stream>>>(B0, Wc[layer], nullptr, B1, N, C, 0);
        // ---- GCN aggregation + self-loop + bias + ReLU -> B0 [N,64]
        k_fill<<<gs(N * HID), 256, 0, stream>>>(B2, 0.0f, N * HID);
        k_edge<<<gs(E * 16), 256, 0, stream>>>(ei, deg, B1, B2, E);
        k_self_relu<<<gs(N * HID), 256, 0, stream>>>(B2, B1, deg, bc[layer], B0, N);
        hsrc = B0;
        C = HID;
    }

    // ---- global mean pool -> gpool [G,64]
    k_fill<<<gs(G * HID), 256, 0, stream>>>(gsum, 0.0f, G * HID);
    k_seg_sum<<<gs(N * HID), 256, 0, stream>>>(B0, batch, gsum, N, HID);
    k_mean<<<gs(G * HID), 256, 0, stream>>>(gsum, gcnt, gpool, G, HID);

    // ---- dense: relu(gpool @ Wd + bd)  (WMMA fp32)
    k_wmma_gemm<<<cdiv_h(G, 128), 256, 0, stream>>>(gpool, Wd, bd, gdense, G, HID, 1);

    // ---- head: logits + softmax -> d_out [G,2]
    k_head<<<1, 256, 0, stream>>>(gdense, Wo, bo, (float*)d_out, G);
}